// CustomWeightedTensorProduct_28518582846293
// MI455X (gfx1250) — compile-verified
//
#include <hip/hip_runtime.h>

// CDNA5 / gfx1250. Memory-bound problem (AI ~0.8 FLOP/B, HBM floor ~24us at
// 23.3 TB/s). Strategy: stream w once, fully coalesced, directly into
// V_WMMA_F32_16X16X4_F32 B-operand lane layout; per-z math = D(16x16) =
// A(16x80 block-diagonal input vectors) x B(80x16 stacked weight matrices),
// accumulated over 20 chained WMMAs.

typedef __attribute__((ext_vector_type(2))) float v2f;
typedef __attribute__((ext_vector_type(8))) float v8f;

#define WAVES_PER_BLOCK 8

__global__ __launch_bounds__(256) void ctp_wmma_kernel(
    const float* __restrict__ x1,   // (Z, 64)
    const float* __restrict__ x2,   // (Z, 4)
    const float* __restrict__ w,    // (Z, 1280)
    float* __restrict__ out,        // (Z, 64)
    int Z)
{
  // Per-wave vector table: 16 K-columns, column stride 20 floats.
  // Column u (written by lane u) holds [v_0[u], v_1[u], ..., v_8[u], 0...].
  // Stride 20 => float4 stores stay 16B-aligned and bank-conflict-free;
  // A-operand reads (lanes 0-15: +m consecutive; lanes 16-31: +40 floats
  // offset) never collide in the 64 LDS banks.
  __shared__ __align__(16) float tbl[WAVES_PER_BLOCK][16 * 20];

  const int lane = threadIdx.x & 31;
  const int wid  = threadIdx.x >> 5;
  const int z    = blockIdx.x * WAVES_PER_BLOCK + wid;
  const int m    = lane & 15;

  float ex = 0.f, ey = 0.f, ez = 0.f;

  if (z < Z) {
    // x2: 16B-aligned broadcast load to all lanes
    const float4 xv = *(const float4*)(x2 + (size_t)z * 4);
    const float s20 = xv.x; ex = xv.y; ey = xv.z; ez = xv.w;

    if (lane < 16) {
      const float* x1z = x1 + (size_t)z * 64;
      const float s10 = x1z[lane];                 // s1_0[u], u = lane
      const float ax  = x1z[16 + 3 * lane + 0];    // s1_1[u, 0..2]
      const float ay  = x1z[16 + 3 * lane + 1];
      const float az  = x1z[16 + 3 * lane + 2];

      // Fold all normalization into the vectors (everything is linear):
      //   out0 rows (0,5): /sqrt(32); out1 rows (1,2-4,6-8): /sqrt(48)
      //   sw3 carries extra 1/sqrt(3), sw4 extra 1/sqrt(2)
      //   1/(sqrt3*sqrt32) == 1/(sqrt2*sqrt48) == 1/sqrt(96)
      const float S32 = 0.17677669529663687f;   // 1/sqrt(32)
      const float S48 = 0.14433756729740643f;   // 1/sqrt(48)
      const float S96 = 0.10206207261596577f;   // 1/sqrt(96)

      const float r0 = s10 * s20 * S32;                 // -> sw0
      const float r1 = s10 * S48;                       // -> sw1 (t1)
      const float r2 = s20 * ax * S48;                  // -> sw2, i=0
      const float r3 = s20 * ay * S48;                  // -> sw2, i=1
      const float r4 = s20 * az * S48;                  // -> sw2, i=2
      const float r5 = (ax * ex + ay * ey + az * ez) * S96;  // -> sw3 (d)
      const float r6 = (ay * ez - az * ey) * S96;       // -> sw4, cross i=0
      const float r7 = (az * ex - ax * ez) * S96;       // -> sw4, cross i=1
      const float r8 = (ax * ey - ay * ex) * S96;       // -> sw4, cross i=2

      float* col = &tbl[wid][lane * 20];
      *(float4*)(col + 0) = make_float4(r0, r1, r2, r3);
      *(float4*)(col + 4) = make_float4(r4, r5, r6, r7);
      *(float4*)(col + 8) = make_float4(r8, 0.f, 0.f, 0.f);
    }
  }
  __syncthreads();
  if (z >= Z) return;

  // Row m of A/D belongs to weight block gm(m); rows 9-15 are zero.
  const int gm = (m == 0) ? 0 : (m == 1) ? 1 : (m <= 4) ? 2
               : (m == 5) ? 3 : (m <= 8) ? 4 : -1;

  const int half = lane >> 4;           // 0: K even base, 1: K odd-half base
  const int u0   = half << 1;           // A layout: VGPR0 = K0|K2, VGPR1 = K1|K3
  const int coff = (half << 5) + m;     // B layout: rows 4j+{0,2} / cols m
  const float* tw = &tbl[wid][0];
  const float* wz = w + (size_t)z * 1280;

  v8f acc = {0.f, 0.f, 0.f, 0.f, 0.f, 0.f, 0.f, 0.f};

#pragma unroll
  for (int g = 0; g < 5; ++g) {
    const bool act = (gm == g);
#pragma unroll
    for (int j = 0; j < 4; ++j) {
      // A operand: block-diagonal gather from LDS vector table (masked)
      float a0 = tw[(u0 + 4 * j    ) * 20 + m];
      float a1 = tw[(u0 + 4 * j + 1) * 20 + m];
      a0 = act ? a0 : 0.f;
      a1 = act ? a1 : 0.f;
      // B operand: weights straight from global, coalesced 128B per load
      const float* bp = wz + g * 256 + j * 64 + coff;
      const float b0 = bp[0];    // rows 4j+0 (lo lanes) / 4j+2 (hi lanes)
      const float b1 = bp[16];   // rows 4j+1 (lo lanes) / 4j+3 (hi lanes)

      v2f A; A.x = a0; A.y = a1;
      v2f B; B.x = b0; B.y = b1;
      // D = A*B + C, chained accumulator
      acc = __builtin_amdgcn_wmma_f32_16x16x4_f32(
          false, A, false, B, (short)0, acc, false, false);
    }
  }

  // D layout: VGPR v, lanes 0-15 -> row v; lanes 16-31 -> row v+8.
  // Row 8 (cross i=2 result) lives in lanes 16-31 of acc[0]: shuffle it down.
  const float r42 = __shfl(acc[0], 16 + m, 32);

  if (lane < 16) {
    float* oz = out + (size_t)z * 64;
    const float t1 = acc[1];                       // (s1_0 @ sw1)/sqrt48
    oz[m] = acc[0] + acc[5];                       // out0 (scales pre-folded)
    oz[16 + 3 * m + 0] = t1 * ex + acc[2] + acc[6];
    oz[16 + 3 * m + 1] = t1 * ey + acc[3] + acc[7];
    oz[16 + 3 * m + 2] = t1 * ez + acc[4] + r42;
  }
}

extern "C" void kernel_launch(void* const* d_in, const int* in_sizes, int n_in,
                              void* d_out, int out_size, void* d_ws, size_t ws_size,
                              hipStream_t stream) {
  const float* x1 = (const float*)d_in[0];
  const float* x2 = (const float*)d_in[1];
  const float* w  = (const float*)d_in[2];
  float* out = (float*)d_out;
  const int Z = in_sizes[0] / 64;
  const int blocks = (Z + WAVES_PER_BLOCK - 1) / WAVES_PER_BLOCK;
  ctp_wmma_kernel<<<blocks, WAVES_PER_BLOCK * 32, 0, stream>>>(x1, x2, w, out, Z);
}